// ContrastiveLoss_17910013624524
// MI455X (gfx1250) — compile-verified
//
#include <hip/hip_runtime.h>

// ---- problem constants (match reference) ----
#define B_SZ   16384
#define D_SZ   128
#define INV_T  5.0f          // 1 / 0.2
#define EPS_F  1e-8f

// ---- launch geometry ----
#define WAVE_SIZE        32              // gfx1250 is wave32
#define WAVES_PER_BLOCK  8
#define BLOCK_THREADS    (WAVES_PER_BLOCK * WAVE_SIZE)   // 256
#define ROWS_PER_WAVE    16              // WMMA M
#define ROWS_PER_BLOCK   (WAVES_PER_BLOCK * ROWS_PER_WAVE) // 128
#define COLS_PER_STAGE   64              // xbar columns staged in LDS per stage
#define KSTEPS           (D_SZ / 4)      // 32 wmma_f32_16x16x4 per 16x16 tile
#define LDS_STRIDE       (D_SZ + 4)      // 132 floats: breaks 64-bank aliasing

typedef __attribute__((ext_vector_type(2))) float v2f;
typedef __attribute__((ext_vector_type(8))) float v8f;

__global__ void zero_acc_kernel(float* acc) {
    acc[0] = 0.0f;
}

__global__ void finalize_kernel(const float* acc, float* out) {
    out[0] = -acc[0] * (1.0f / (float)B_SZ);
}

__global__ __launch_bounds__(BLOCK_THREADS)
void contrastive_main(const float* __restrict__ x,
                      const float* __restrict__ xbar,
                      float* __restrict__ acc) {
    __shared__ __align__(16) float lds_xbar[COLS_PER_STAGE * LDS_STRIDE];

    const int tid  = threadIdx.x;
    const int wave = tid >> 5;
    const int lane = tid & 31;
    const int half = lane >> 4;      // which 16-lane half of the wave
    const int n    = lane & 15;      // N index inside tile (also M index for A rows)

    const int row0 = blockIdx.x * ROWS_PER_BLOCK + wave * ROWS_PER_WAVE;

    // ---- preload A fragments: x rows for this wave, kept in VGPRs all kernel ----
    // A 16x4 f32 layout: lane = M, VGPR v holds K = v + 2*half  -> 2 consecutive
    // floats per lane per k-step at offset ks*4 + 2*half.
    v2f afrag[KSTEPS];
    {
        const float* xrow = x + (size_t)(row0 + n) * D_SZ + 2 * half;
        #pragma unroll
        for (int ks = 0; ks < KSTEPS; ++ks)
            afrag[ks] = *(const v2f*)(xrow + ks * 4);
    }

    float negsum[8];
    #pragma unroll
    for (int r = 0; r < 8; ++r) negsum[r] = 0.0f;
    float diag = 0.0f;   // one diagonal dot captured per "owning" lane

    for (int cs = 0; cs < B_SZ; cs += COLS_PER_STAGE) {
        __syncthreads();   // previous stage's LDS reads done
        // ---- cooperative stage of xbar[cs .. cs+63][:] into padded LDS ----
        {
            const int lrow = tid >> 2;          // 0..63
            const int lq   = tid & 3;           // quarter of the 128-float row
            const float4* src =
                (const float4*)(xbar + (size_t)(cs + lrow) * D_SZ + lq * 32);
            float* dstbase = lds_xbar + lrow * LDS_STRIDE + lq * 32;
            #pragma unroll
            for (int i = 0; i < 8; ++i)
                *(float4*)(dstbase + i * 4) = src[i];
        }
        __syncthreads();   // stage visible to all waves

        // ---- prefetch next stage (1 x 128B line per thread) to hide HBM/L2
        //      latency behind the 128 WMMAs of this stage ----
        if (cs + COLS_PER_STAGE < B_SZ) {
            __builtin_prefetch(
                xbar + (size_t)(cs + COLS_PER_STAGE) * D_SZ + (size_t)tid * 32,
                0 /*read*/, 3);
        }

        #pragma unroll
        for (int sub = 0; sub < COLS_PER_STAGE / 16; ++sub) {
            const int j0 = cs + sub * 16;
            // B 4x16 f32 layout mirrors A: lane = N (column j), VGPR v holds
            // K = v + 2*half.  Preload the whole 16x128 B tile into registers
            // so the WMMA chain runs without per-iteration dscnt stalls.
            const float* brow =
                lds_xbar + (sub * 16 + n) * LDS_STRIDE + 2 * half;
            v2f bfrag[KSTEPS];
            #pragma unroll
            for (int ks = 0; ks < KSTEPS; ++ks)
                bfrag[ks] = *(const v2f*)(brow + ks * 4);

            v8f acc0 = {};   // four independent accumulation chains
            v8f acc1 = {};
            v8f acc2 = {};
            v8f acc3 = {};
            #pragma unroll
            for (int ks = 0; ks < KSTEPS; ks += 4) {
                acc0 = __builtin_amdgcn_wmma_f32_16x16x4_f32(
                        false, afrag[ks + 0], false, bfrag[ks + 0], (short)0, acc0, false, false);
                acc1 = __builtin_amdgcn_wmma_f32_16x16x4_f32(
                        false, afrag[ks + 1], false, bfrag[ks + 1], (short)0, acc1, false, false);
                acc2 = __builtin_amdgcn_wmma_f32_16x16x4_f32(
                        false, afrag[ks + 2], false, bfrag[ks + 2], (short)0, acc2, false, false);
                acc3 = __builtin_amdgcn_wmma_f32_16x16x4_f32(
                        false, afrag[ks + 3], false, bfrag[ks + 3], (short)0, acc3, false, false);
            }

            // C/D layout: lane L holds column N=L%16, rows M = r + 8*(L/16).
            float simv[8];
            #pragma unroll
            for (int r = 0; r < 8; ++r)
                simv[r] = (acc0[r] + acc1[r]) + (acc2[r] + acc3[r]);

            if (j0 == row0) {          // wave-uniform: tile crossing the diagonal
                #pragma unroll
                for (int r = 0; r < 8; ++r)
                    if (n == r + 8 * half) diag = simv[r];
            }
            #pragma unroll
            for (int r = 0; r < 8; ++r)
                negsum[r] += __expf(simv[r] * INV_T);
        }
    }

    // ---- per-row reduction: sum over 16 lanes sharing the same rows ----
    float termAcc = 0.0f;
    #pragma unroll
    for (int r = 0; r < 8; ++r) {
        float srow = negsum[r];
        srow += __shfl_xor(srow, 1, WAVE_SIZE);
        srow += __shfl_xor(srow, 2, WAVE_SIZE);
        srow += __shfl_xor(srow, 4, WAVE_SIZE);
        srow += __shfl_xor(srow, 8, WAVE_SIZE);
        // row (row0 + r + 8*half)'s term computed in its owning lane
        if (n == r + 8 * half) {
            float pos = __expf(diag * INV_T);
            termAcc = __logf(pos / (srow + EPS_F));   // log(pos/(neg+eps))
        }
    }
    // ---- wave reduce the 16 row terms, one atomic per wave ----
    #pragma unroll
    for (int off = 16; off >= 1; off >>= 1)
        termAcc += __shfl_xor(termAcc, off, WAVE_SIZE);
    if (lane == 0)
        atomicAdd(acc, termAcc);
}

extern "C" void kernel_launch(void* const* d_in, const int* in_sizes, int n_in,
                              void* d_out, int out_size, void* d_ws, size_t ws_size,
                              hipStream_t stream) {
    const float* x    = (const float*)d_in[0];
    const float* xbar = (const float*)d_in[1];
    float* out = (float*)d_out;
    float* acc = (float*)d_ws;

    zero_acc_kernel<<<1, 1, 0, stream>>>(acc);
    contrastive_main<<<B_SZ / ROWS_PER_BLOCK, BLOCK_THREADS, 0, stream>>>(x, xbar, acc);
    finalize_kernel<<<1, 1, 0, stream>>>(acc, out);
}